// AblationGNNAutoencoder_52158082842630
// MI455X (gfx1250) — compile-verified
//
#include <hip/hip_runtime.h>
#include <hip/hip_bf16.h>

typedef __attribute__((ext_vector_type(2))) float v2f;
typedef __attribute__((ext_vector_type(8))) float v8f;

#define TILE_M 128
#define TILE_N 128
#define TILE_K 16
#define XS_STRIDE 20   // 16 cols + 4 pad -> 16B-aligned float4 stores, spread banks
#define WS_STRIDE 132  // 128 cols + 4 pad

// ---------------------------------------------------------------------------
// Dense GEMM: C[M,N] = X[M,K] * W[K,N], fp32, using V_WMMA_F32_16X16X4_F32.
// 256 threads = 8 waves; each wave owns a 32x64 sub-tile (2x4 16x16 accums).
// ---------------------------------------------------------------------------
__global__ __launch_bounds__(256)
void gemm_f32_wmma(const float* __restrict__ X, const float* __restrict__ W,
                   float* __restrict__ C, int M, int N, int K)
{
    __shared__ float Xs[TILE_M * XS_STRIDE];
    __shared__ float Ws[TILE_K * WS_STRIDE];

    const int tid  = threadIdx.x;
    const int wave = tid >> 5;
    const int lane = tid & 31;
    const int lm   = lane & 15;          // row (A) / col (B) within 16
    const int lk2  = (lane >> 4) << 1;   // k-pair base: 0 or 2 (ISA 7.12.2)
    const int half = lane >> 4;          // C/D: M = vgpr + 8*half

    const int wm = wave >> 1;            // 0..3 -> M offset wm*32
    const int wn = wave & 1;             // 0..1 -> N offset wn*64

    const int block_m = blockIdx.y * TILE_M;
    const int block_n = blockIdx.x * TILE_N;

    v8f acc[2][4];
#pragma unroll
    for (int mt = 0; mt < 2; ++mt)
#pragma unroll
        for (int nt = 0; nt < 4; ++nt)
            acc[mt][nt] = (v8f){0.f, 0.f, 0.f, 0.f, 0.f, 0.f, 0.f, 0.f};

    // cooperative staging indices
    const int xr = tid >> 1;            // X tile: 128 rows x 16 cols, 8 floats/thread
    const int xc = (tid & 1) * 8;
    const int wr = tid >> 4;            // W tile: 16 rows x 128 cols, 8 floats/thread
    const int wc = (tid & 15) * 8;

    for (int k0 = 0; k0 < K; k0 += TILE_K) {
        // ---- stage X tile (zero-fill beyond M) ----
        {
            const int gr = block_m + xr;
            float4 a0 = make_float4(0.f, 0.f, 0.f, 0.f);
            float4 a1 = make_float4(0.f, 0.f, 0.f, 0.f);
            if (gr < M) {
                const float* src = X + (size_t)gr * K + (k0 + xc);
                a0 = *(const float4*)(src);
                a1 = *(const float4*)(src + 4);
            }
            *(float4*)&Xs[xr * XS_STRIDE + xc]     = a0;
            *(float4*)&Xs[xr * XS_STRIDE + xc + 4] = a1;
        }
        // ---- stage W tile (K multiple of 16, N multiple of 128: no guards) ----
        {
            const float* src = W + (size_t)(k0 + wr) * N + (block_n + wc);
            float4 b0 = *(const float4*)(src);
            float4 b1 = *(const float4*)(src + 4);
            *(float4*)&Ws[wr * WS_STRIDE + wc]     = b0;
            *(float4*)&Ws[wr * WS_STRIDE + wc + 4] = b1;
        }
        __syncthreads();

#pragma unroll
        for (int ks = 0; ks < TILE_K / 4; ++ks) {
            const int kk = ks * 4 + lk2;
            // A fragments: lane holds X[m, kk], X[m, kk+1]
            v2f afrag[2];
            afrag[0] = *(const v2f*)&Xs[(wm * 32 + lm) * XS_STRIDE + kk];
            afrag[1] = *(const v2f*)&Xs[(wm * 32 + 16 + lm) * XS_STRIDE + kk];
            // B fragments: lane holds W[kk, n], W[kk+1, n]
            v2f bfrag[4];
#pragma unroll
            for (int nt = 0; nt < 4; ++nt) {
                const int col = wn * 64 + nt * 16 + lm;
                v2f b;
                b.x = Ws[kk * WS_STRIDE + col];
                b.y = Ws[(kk + 1) * WS_STRIDE + col];
                bfrag[nt] = b;
            }
#pragma unroll
            for (int mt = 0; mt < 2; ++mt)
#pragma unroll
                for (int nt = 0; nt < 4; ++nt)
                    acc[mt][nt] = __builtin_amdgcn_wmma_f32_16x16x4_f32(
                        false, afrag[mt], false, bfrag[nt],
                        (short)0, acc[mt][nt], false, false);
        }
        __syncthreads();
    }

    // ---- store C; lane half h holds rows r+8h of each 16x16 tile ----
#pragma unroll
    for (int mt = 0; mt < 2; ++mt) {
        const int mbase = block_m + wm * 32 + mt * 16 + half * 8;
        const int nbase = block_n + wn * 64 + lm;
#pragma unroll
        for (int r = 0; r < 8; ++r) {
            const int row = mbase + r;
            if (row < M) {
#pragma unroll
                for (int nt = 0; nt < 4; ++nt)
                    C[(size_t)row * N + nbase + nt * 16] = acc[mt][nt][r];
            }
        }
    }
}

// ---------------------------------------------------------------------------
// Graph preprocessing kernels (run once, reused by all 4 layers)
// ---------------------------------------------------------------------------
__global__ void fill_zero(float* __restrict__ p, long long n)
{
    long long i = (long long)blockIdx.x * blockDim.x + threadIdx.x;
    if (i < n) p[i] = 0.f;
}

__global__ void deg_accum(const int* __restrict__ dst, const float* __restrict__ ew,
                          float* __restrict__ deg, int nE)
{
    int e = blockIdx.x * blockDim.x + threadIdx.x;
    if (e < nE) atomicAdd(&deg[dst[e]], ew[e]);
}

__global__ void deg_to_dinv(float* __restrict__ d, int n)
{
    int i = blockIdx.x * blockDim.x + threadIdx.x;
    if (i < n) d[i] = rsqrtf(d[i] + 1.0f);   // self-loop weight 1; deg+1 > 0 always
}

__global__ void edge_norm(const int* __restrict__ src, const int* __restrict__ dst,
                          const float* __restrict__ ew, const float* __restrict__ dinv,
                          float* __restrict__ nrm, int nE)
{
    int e = blockIdx.x * blockDim.x + threadIdx.x;
    if (e < nE) nrm[e] = dinv[src[e]] * ew[e] * dinv[dst[e]];
}

// ---------------------------------------------------------------------------
// SpMM scatter: agg[dst] += norm[e] * xw[src]; (N/4) threads per edge, float4
// gathers (coalesced: a wave covers consecutive features of one edge's row).
// ---------------------------------------------------------------------------
__global__ void spmm_scatter(const int* __restrict__ src, const int* __restrict__ dst,
                             const float* __restrict__ nrm, const float* __restrict__ xw,
                             float* __restrict__ agg, int nE, int N, int shift)
{
    long long gid = (long long)blockIdx.x * blockDim.x + threadIdx.x;
    int e = (int)(gid >> shift);
    if (e >= nE) return;
    int f = (int)(gid & ((1 << shift) - 1)) << 2;
    float w = nrm[e];
    const float4 v = *(const float4*)(xw + (size_t)src[e] * N + f);
    float* o = agg + (size_t)dst[e] * N + f;
    atomicAdd(o + 0, w * v.x);
    atomicAdd(o + 1, w * v.y);
    atomicAdd(o + 2, w * v.z);
    atomicAdd(o + 3, w * v.w);
}

// out[i,f] = (agg[i,f] + dinv[i]^2 * xw[i,f] + b[f])  (optionally ReLU'd)
__global__ void layer_finish(float* __restrict__ out, const float* __restrict__ xw,
                             const float* __restrict__ dinv, const float* __restrict__ bias,
                             long long total, int logN, int relu)
{
    long long i = (long long)blockIdx.x * blockDim.x + threadIdx.x;
    if (i >= total) return;
    int col = (int)(i & ((1LL << logN) - 1));
    int row = (int)(i >> logN);
    float di = dinv[row];
    float v = out[i] + di * di * xw[i] + bias[col];
    out[i] = relu ? fmaxf(v, 0.f) : v;
}

// ---------------------------------------------------------------------------
// Host launcher
// ---------------------------------------------------------------------------
extern "C" void kernel_launch(void* const* d_in, const int* in_sizes, int n_in,
                              void* d_out, int out_size, void* d_ws, size_t ws_size,
                              hipStream_t stream)
{
    const float* x  = (const float*)d_in[0];
    const int*   ei = (const int*)d_in[1];     // [2][E]: row0 = src, row1 = dst
    const float* ew = (const float*)d_in[2];
    const float* W1 = (const float*)d_in[3];  const float* b1 = (const float*)d_in[4];
    const float* W2 = (const float*)d_in[5];  const float* b2 = (const float*)d_in[6];
    const float* W3 = (const float*)d_in[7];  const float* b3 = (const float*)d_in[8];
    const float* W4 = (const float*)d_in[9];  const float* b4 = (const float*)d_in[10];

    const int M = in_sizes[0] / 512;   // 50000 nodes
    const int E = in_sizes[2];         // 800000 edges
    const int* src = ei;
    const int* dst = ei + E;

    // workspace carve-up (fp32): dinv | norm | xw (M x 512)
    float* dinv = (float*)d_ws;
    float* nrm  = dinv + (((M + 511) / 512) * 512);
    float* xw   = nrm  + (((E + 255) / 256) * 256);

    float* out  = (float*)d_out;
    float* hbuf = out;                      // reused: h, then h2, then x_recon
    float* zbuf = out + (size_t)M * 512;    // z output region (M x 128)

    const int T = 256;

    // ---- graph preprocessing (once) ----
    fill_zero  <<<(M + T - 1) / T, T, 0, stream>>>(dinv, M);
    deg_accum  <<<(E + T - 1) / T, T, 0, stream>>>(dst, ew, dinv, E);
    deg_to_dinv<<<(M + T - 1) / T, T, 0, stream>>>(dinv, M);
    edge_norm  <<<(E + T - 1) / T, T, 0, stream>>>(src, dst, ew, dinv, nrm, E);

    auto run_layer = [&](const float* Xin, const float* W, const float* b,
                         float* outb, int K, int N, int relu, int logN) {
        dim3 grid(N / TILE_N, (M + TILE_M - 1) / TILE_M);
        gemm_f32_wmma<<<grid, T, 0, stream>>>(Xin, W, xw, M, N, K);

        long long total = (long long)M * N;
        fill_zero<<<(unsigned)((total + T - 1) / T), T, 0, stream>>>(outb, total);

        int shift = logN - 2;                       // (N/4) threads per edge
        long long nthr = (long long)E << shift;
        spmm_scatter<<<(unsigned)((nthr + T - 1) / T), T, 0, stream>>>(
            src, dst, nrm, xw, outb, E, N, shift);

        layer_finish<<<(unsigned)((total + T - 1) / T), T, 0, stream>>>(
            outb, xw, dinv, b, total, logN, relu);
    };

    // encode
    run_layer(x,    W1, b1, hbuf, 512, 512, /*relu=*/1, 9);   // h  (in x_recon slot)
    run_layer(hbuf, W2, b2, zbuf, 512, 128, /*relu=*/0, 7);   // z  (final output)
    // decode
    run_layer(zbuf, W3, b3, hbuf, 128, 512, /*relu=*/1, 9);   // h2 (overwrites h)
    run_layer(hbuf, W4, b4, hbuf, 512, 512, /*relu=*/0, 9);   // x_recon (GEMM reads
                                                              // h2 before zero-fill)
}